// FermiNet_spin_6244882448428
// MI455X (gfx1250) — compile-verified
//
#include <hip/hip_runtime.h>
#include <hip/hip_bf16.h>

// Problem constants from the reference
#define NPART 2048
#define NDIM  3
#define NBATCH 8
#define BLOCK 256
// L = (2048 / 0.016355)^(1/3) computed in double, then used as f32 (matches JAX cast)
#define BOX_L   50.0295352f
#define INV_L   (1.0f / 50.0295352f)
#define B_DECAY 2.6f

typedef unsigned int u32;
typedef unsigned long long u64;
typedef __attribute__((ext_vector_type(4))) unsigned int v4u;
typedef __attribute__((ext_vector_type(8))) int          v8i;
typedef __attribute__((ext_vector_type(4))) int          v4i;

__global__ __launch_bounds__(BLOCK) void backflow_kernel(const float* __restrict__ x,
                                                         float* __restrict__ out) {
    // Batch slab of positions staged in LDS: 2048 * 3 * 4B = 24 KB (of 320 KB/WGP)
    __shared__ float xs[NPART * NDIM];

    const int b = blockIdx.y;
    const int i = blockIdx.x * BLOCK + (int)threadIdx.x;
    const float* xb = x + (size_t)b * NPART * NDIM;

#if __has_builtin(__builtin_amdgcn_tensor_load_to_lds)
    // --- TDM: DMA the whole batch slab (6144 f32) global -> LDS, one issue per WG ---
    // Scalar (SGPR) branch so only wave 0 reaches the TDM instruction: TDM issues
    // regardless of EXEC, so an EXEC-predicated guard would issue it 8 times.
    if (__builtin_amdgcn_readfirstlane(threadIdx.x) == 0u) {
        // Flat LDS pointer: aperture occupies bits [63:32]; bits [31:0] are the
        // LDS byte offset (ISA 10.2), which is what D#.lds_addr wants.
        u32 lds_off = (u32)(size_t)(&xs[0]);
        u64 ga = (u64)(size_t)xb;

        // D# group 0: count=1 (valid), lds_addr, global_addr[56:0], type=2
        v4u g0;
        g0[0] = 1u;                                   // count=1, is_restore=0, gather=0
        g0[1] = lds_off;                              // lds_addr (bytes)
        g0[2] = (u32)ga;                              // global_addr[31:0]
        g0[3] = ((u32)(ga >> 32) & 0x01FFFFFFu)       // global_addr[56:32]
              | 0x80000000u;                          // type=2 at bits[127:126]

        // D# group 1: data_size=4B; tensor_dim0 = tile_dim0 = 6144; 1 row.
        v8i g1;
        g1[0] = (int)(2u << 16);                      // workgroup_mask=0, data_size=2 (4B)
        g1[1] = (int)(6144u << 16);                   // tensor_dim0[15:0] << 16 (low half)
        g1[2] = (int)(1u << 16);                      // tensor_dim0 hi=0, tensor_dim1=1
        g1[3] = (int)(6144u << 16);                   // tensor_dim1 hi=0, tile_dim0=6144
        g1[4] = 1;                                    // tile_dim1=1, tile_dim2=0 (unused)
        g1[5] = 6144;                                 // tensor_dim0_stride[31:0]
        g1[6] = 0;                                    // stride hi, tensor_dim1_stride lo
        g1[7] = 0;                                    // tensor_dim1_stride hi

        v4i gz4 = {0, 0, 0, 0};                       // groups 2/3: dims 2..4 unused
        v8i gz8 = {0, 0, 0, 0, 0, 0, 0, 0};           // trailing unused group (clang-23 form)

        __builtin_amdgcn_tensor_load_to_lds(g0, g1, gz4, gz4, gz8, /*cpol=*/0);
        __builtin_amdgcn_s_wait_tensorcnt(0);         // s_wait_tensorcnt 0
    }
#else
    // Fallback: cooperative fill (24 elements per thread)
    for (int k = threadIdx.x; k < NPART * NDIM; k += BLOCK) xs[k] = xb[k];
#endif
    __syncthreads();

    // Own (raw) position; wrap only for the output base term. The minimum-image
    // step makes the pairwise part invariant to shifts by exact multiples of L.
    const float xi0 = xb[i * 3 + 0];
    const float xi1 = xb[i * 3 + 1];
    const float xi2 = xb[i * 3 + 2];

    float a0 = 0.0f, a1 = 0.0f, a2 = 0.0f;

#pragma unroll 4
    for (int j = 0; j < NPART; ++j) {
        // Uniform-address LDS reads: hardware broadcast, no bank conflicts.
        float d0 = xi0 - xs[j * 3 + 0];
        float d1 = xi1 - xs[j * 3 + 1];
        float d2 = xi2 - xs[j * 3 + 2];

        // Minimum image: d -= L * rint(d / L)   (v_rndne_f32, round-half-even)
        d0 -= BOX_L * __builtin_rintf(d0 * INV_L);
        d1 -= BOX_L * __builtin_rintf(d1 * INV_L);
        d2 -= BOX_L * __builtin_rintf(d2 * INV_L);

        float r2   = d0 * d0 + d1 * d1 + d2 * d2;
        float rinv = __builtin_amdgcn_rsqf(r2);       // v_rsq_f32; +inf on self-pair
        float u    = B_DECAY * rinv;                  // (B/d)
        float u2   = u * u;
        float u4   = u2 * u2;
        float u5   = u4 * u;                          // (B/d)^5
        float e    = __expf(-u5);                     // v_exp_f32; exp(-inf)=0 on self
        float fr   = 0.5f * e - 0.5f;                 // self-pair: fr=-0.5 but rij=0

        a0 = fmaf(fr, d0, a0);
        a1 = fmaf(fr, d1, a1);
        a2 = fmaf(fr, d2, a2);
    }

    // Wrapped base position (identity for x in [0,L), but matches reference)
    float w0 = xi0 - BOX_L * __builtin_floorf(xi0 * INV_L);
    float w1 = xi1 - BOX_L * __builtin_floorf(xi1 * INV_L);
    float w2 = xi2 - BOX_L * __builtin_floorf(xi2 * INV_L);

    float* o = out + ((size_t)b * NPART + (size_t)i) * NDIM;
    o[0] = w0 + a0;
    o[1] = w1 + a1;
    o[2] = w2 + a2;
}

extern "C" void kernel_launch(void* const* d_in, const int* in_sizes, int n_in,
                              void* d_out, int out_size, void* d_ws, size_t ws_size,
                              hipStream_t stream) {
    (void)in_sizes; (void)n_in; (void)d_ws; (void)ws_size; (void)out_size;
    const float* x = (const float*)d_in[0];
    float* out = (float*)d_out;

    dim3 grid(NPART / BLOCK, NBATCH);   // (8, 8) workgroups
    dim3 block(BLOCK);                  // 256 threads = 8 wave32
    backflow_kernel<<<grid, block, 0, stream>>>(x, out);
}